// Involution2d_40286793236507
// MI455X (gfx1250) — compile-verified
//
#include <hip/hip_runtime.h>

// ---------------------------------------------------------------------------
// Involution2d for MI455X (gfx1250, wave32).
//   Phase 1: per-pixel kernel-generating MLP as WMMA f32 GEMMs.
//   Phase 2: bandwidth-bound 7x7 involution apply (L2-resident x).
// ---------------------------------------------------------------------------

typedef __attribute__((ext_vector_type(2))) float v2f;
typedef __attribute__((ext_vector_type(8))) float v8f;

#define B_   4
#define C_   128
#define CR_  32
#define H_   64
#define W_   64
#define HW_  (H_ * W_)
#define K_   7
#define KK_  49          // K*K
#define PAD_ 3
#define EPS_ 1e-5f

// ===========================================================================
// Kernel A: generate Ker[B, 49, H, W] with WMMA.
// One wave handles one tile of 16 consecutive pixels (N dimension).
//   GEMM1: F[32,16]  = W1[32,128]  * X[128,16]   (32 K-steps, 2 M-tiles)
//   epilogue: +b1, BN(eval), ReLU  -> stage to LDS
//   GEMM2: D[64,16]  = W2pad[64,32]* F[32,16]    (8 K-steps, 4 M-tiles, rows>=49 zero)
//   +b2, store rows 0..48.
// Fragment layouts (ISA 7.12.2, f32 16x16x4):
//   A: lane L holds A[M=L%16, K=2*(L/16)+v]   (v = component 0/1)
//   B: lane L holds B[K=2*(L/16)+v, N=L%16]
//   C/D: vgpr r, lane L -> D[M=r+8*(L/16), N=L%16]
// ===========================================================================
__global__ __launch_bounds__(256) void gen_kernel_wmma(
    const float* __restrict__ x,
    const float* __restrict__ w1, const float* __restrict__ b1,
    const float* __restrict__ bn_gamma, const float* __restrict__ bn_beta,
    const float* __restrict__ bn_mean,  const float* __restrict__ bn_var,
    const float* __restrict__ w2, const float* __restrict__ b2,
    float* __restrict__ ker)
{
    __shared__ float lds_f[8][CR_][16];        // per-wave F staging: 16 KB

    const int lane  = threadIdx.x & 31;
    const int wave  = threadIdx.x >> 5;
    const int tile  = blockIdx.x * 8 + wave;   // 16-pixel tile id (1024 total)
    const int p0    = tile * 16;
    const int b     = p0 / HW_;                // HW_ % 16 == 0 -> tiles never cross images
    const int hw0   = p0 % HW_;

    const int nm    = lane & 15;               // N (pixel) or M (row) index within tile
    const int khalf = lane >> 4;               // 0/1 -> K-pair base 0/2

    const float* __restrict__ xb = x + (size_t)b * C_ * HW_;

    // -------------------- GEMM1: F = W1 * X --------------------
    v8f acc0 = {};   // channels 0..15
    v8f acc1 = {};   // channels 16..31
    for (int kk = 0; kk < C_ / 4; ++kk) {
        const int kb = kk * 4 + khalf * 2;     // this lane's K base (2 values)
        v2f bf;                                 // B frag: X[kb..kb+1, pixel nm]
        bf.x = xb[(size_t)(kb + 0) * HW_ + hw0 + nm];
        bf.y = xb[(size_t)(kb + 1) * HW_ + hw0 + nm];
        v2f a0, a1;                             // A frags: W1[row, kb..kb+1]
        a0.x = w1[(nm      ) * C_ + kb + 0];
        a0.y = w1[(nm      ) * C_ + kb + 1];
        a1.x = w1[(nm + 16 ) * C_ + kb + 0];
        a1.y = w1[(nm + 16 ) * C_ + kb + 1];
        acc0 = __builtin_amdgcn_wmma_f32_16x16x4_f32(false, a0, false, bf,
                                                     (short)0, acc0, false, false);
        acc1 = __builtin_amdgcn_wmma_f32_16x16x4_f32(false, a1, false, bf,
                                                     (short)0, acc1, false, false);
    }

    // -------------------- epilogue: +b1, BN, ReLU -> LDS --------------------
#pragma unroll
    for (int r = 0; r < 8; ++r) {
        const int ch0 = r + khalf * 8;         // acc0 row -> channel
        float s0 = bn_gamma[ch0] * rsqrtf(bn_var[ch0] + EPS_);
        float f0 = (acc0[r] + b1[ch0]) * s0 + (bn_beta[ch0] - bn_mean[ch0] * s0);
        lds_f[wave][ch0][nm] = fmaxf(f0, 0.0f);

        const int ch1 = 16 + r + khalf * 8;    // acc1 row -> channel
        float s1 = bn_gamma[ch1] * rsqrtf(bn_var[ch1] + EPS_);
        float f1 = (acc1[r] + b1[ch1]) * s1 + (bn_beta[ch1] - bn_mean[ch1] * s1);
        lds_f[wave][ch1][nm] = fmaxf(f1, 0.0f);
    }
    // Per-wave LDS region; DS ops are in-order within a wave, but fence the
    // compiler and the DS counter before re-reading.
    asm volatile("s_wait_dscnt 0" ::: "memory");

    // -------------------- GEMM2: D = W2pad * F --------------------
    v8f d0 = {}, d1 = {}, d2 = {}, d3 = {};
    for (int kk = 0; kk < CR_ / 4; ++kk) {     // 8 K-steps
        const int kb = kk * 4 + khalf * 2;
        v2f bf;                                 // B frag: F[kb..kb+1, pixel nm]
        bf.x = lds_f[wave][kb + 0][nm];
        bf.y = lds_f[wave][kb + 1][nm];
#pragma unroll
        for (int mt = 0; mt < 4; ++mt) {
            const int row = mt * 16 + nm;       // padded output row (0..63)
            const int rr  = row < KK_ ? row : (KK_ - 1);   // clamp: no OOB load
            v2f a;
            a.x = (row < KK_) ? w2[rr * CR_ + kb + 0] : 0.0f;
            a.y = (row < KK_) ? w2[rr * CR_ + kb + 1] : 0.0f;
            v8f* dd = (mt == 0) ? &d0 : (mt == 1) ? &d1 : (mt == 2) ? &d2 : &d3;
            *dd = __builtin_amdgcn_wmma_f32_16x16x4_f32(false, a, false, bf,
                                                        (short)0, *dd, false, false);
        }
    }

    // -------------------- +b2, store Ker rows 0..48 --------------------
    float* __restrict__ kerb = ker + (size_t)b * KK_ * HW_;
#pragma unroll
    for (int mt = 0; mt < 4; ++mt) {
        const v8f& dd = (mt == 0) ? d0 : (mt == 1) ? d1 : (mt == 2) ? d2 : d3;
#pragma unroll
        for (int r = 0; r < 8; ++r) {
            const int row = mt * 16 + r + khalf * 8;
            if (row < KK_) {
                kerb[(size_t)row * HW_ + hw0 + nm] = dd[r] + b2[row];
            }
        }
    }
}

// ===========================================================================
// Kernel B: involution apply. Block = one (b, h) row; 49x64 kernel slab in
// LDS shared by all 128 channels; each thread streams 32 outputs.
// x (8.4 MB) is L2-resident, so the 49-tap re-reads stay on-chip.
// ===========================================================================
__global__ __launch_bounds__(256) void involution_apply(
    const float* __restrict__ x,
    const float* __restrict__ ker,
    float* __restrict__ out)
{
    __shared__ float kl[KK_][W_];              // 12.25 KB

    const int h = blockIdx.x;
    const int b = blockIdx.y;

    const float* __restrict__ kerb = ker + (size_t)b * KK_ * HW_ + (size_t)h * W_;
    for (int idx = threadIdx.x; idx < KK_ * W_; idx += 256) {
        const int ij = idx / W_;
        const int w  = idx - ij * W_;
        kl[ij][w] = kerb[(size_t)ij * HW_ + w];
    }
    __syncthreads();

    const int w  = threadIdx.x & (W_ - 1);     // 0..63
    const int cb = threadIdx.x >> 6;           // 0..3

    const float* __restrict__ xb = x   + (size_t)b * C_ * HW_;
    float* __restrict__       ob = out + (size_t)b * C_ * HW_ + (size_t)h * W_;

    for (int c = cb; c < C_; c += 4) {
        const float* __restrict__ xc = xb + (size_t)c * HW_;
        float acc = 0.0f;
#pragma unroll
        for (int i = 0; i < K_; ++i) {
            const int hh = h + i - PAD_;
            if (hh < 0 || hh >= H_) continue;
#pragma unroll
            for (int j = 0; j < K_; ++j) {
                const int ww = w + j - PAD_;
                if (ww < 0 || ww >= W_) continue;
                acc = fmaf(kl[i * K_ + j][w], xc[(size_t)hh * W_ + ww], acc);
            }
        }
        ob[(size_t)c * HW_ + w] = acc;
    }
}

// ===========================================================================
extern "C" void kernel_launch(void* const* d_in, const int* in_sizes, int n_in,
                              void* d_out, int out_size, void* d_ws, size_t ws_size,
                              hipStream_t stream) {
    const float* x        = (const float*)d_in[0];
    const float* w1       = (const float*)d_in[1];
    const float* b1       = (const float*)d_in[2];
    const float* bn_gamma = (const float*)d_in[3];
    const float* bn_beta  = (const float*)d_in[4];
    const float* bn_mean  = (const float*)d_in[5];
    const float* bn_var   = (const float*)d_in[6];
    const float* w2       = (const float*)d_in[7];
    const float* b2       = (const float*)d_in[8];
    float* out = (float*)d_out;
    float* ker = (float*)d_ws;   // B * 49 * H * W floats = 3.2 MB scratch

    // Phase 1: 1024 pixel-tiles, 8 waves per 256-thread block -> 128 blocks.
    const int tiles = (B_ * HW_) / 16;
    gen_kernel_wmma<<<dim3(tiles / 8), dim3(256), 0, stream>>>(
        x, w1, b1, bn_gamma, bn_beta, bn_mean, bn_var, w2, b2, ker);

    // Phase 2: one block per (h, b) row.
    involution_apply<<<dim3(H_, B_), dim3(256), 0, stream>>>(x, ker, out);
}